// HeteroGraphSage_50749333570054
// MI455X (gfx1250) — compile-verified
//
#include <hip/hip_runtime.h>
#include <hip/hip_bf16.h>

// ---------------- problem constants (fixed by the reference) ----------------
constexpr int kH   = 256;     // hidden
constexpr int kOUT = 128;     // out channels
constexpr int kNC  = 20000;   // central nodes
constexpr int kN1  = 60000;   // child_cont nodes
constexpr int kN2  = 60000;   // child_categ nodes
constexpr int kE   = 200000;  // edges per relation
constexpr int kB   = 512;     // graphs

typedef __attribute__((ext_vector_type(2))) float v2f;
typedef __attribute__((ext_vector_type(8))) float v8f;

// ---------------------------------------------------------------------------
// Fused multi-operand f32 WMMA GEMM, compile-time K and N so all B-tile
// strides fold into global_load immediate offsets:
//   C[M x N] = maybe_relu( out_scale * sum_p (A_p[M x K] @ W_p[K x N])
//                          + bias_scale * (biasA[n] + biasB[n]) )
// Block = 256 threads = 8 waves; wave w computes C tile
//   rows [16*blockIdx.x, +16), cols [16*(blockIdx.y*8+w), +16).
// Requires: M % 16 == 0, K % 4 == 0, N % 128 == 0 (all hold here).
// Unused pairs / biases are passed as nullptr.
// ---------------------------------------------------------------------------
template <int K, int N>
__global__ __launch_bounds__(256) void gemm_wmma_f32(
    const float* __restrict__ A0, const float* __restrict__ W0,
    const float* __restrict__ A1, const float* __restrict__ W1,
    const float* __restrict__ A2, const float* __restrict__ W2,
    const float* __restrict__ A3, const float* __restrict__ W3,
    const float* __restrict__ biasA, const float* __restrict__ biasB,
    float bias_scale, float out_scale, int do_relu,
    float* __restrict__ C)
{
    const int lane  = threadIdx.x & 31;
    const int wave  = threadIdx.x >> 5;
    const int row0  = blockIdx.x * 16;
    const int col0  = (blockIdx.y * 8 + wave) * 16;
    const int half  = lane >> 4;   // 0: lanes 0-15, 1: lanes 16-31
    const int l16   = lane & 15;

    v8f acc = {0.f, 0.f, 0.f, 0.f, 0.f, 0.f, 0.f, 0.f};

#pragma unroll
    for (int p = 0; p < 4; ++p) {
        const float* A = (p == 0) ? A0 : (p == 1) ? A1 : (p == 2) ? A2 : A3;
        const float* W = (p == 0) ? W0 : (p == 1) ? W1 : (p == 2) ? W2 : W3;
        if (A == nullptr) continue;
        // A fragment: lane -> (m = row0+l16, k = k0 + half*2 + {0,1})
        const float* arow = A + (size_t)(row0 + l16) * K + half * 2;
        // B fragment: lane -> (k = k0 + half*2 + v, n = col0 + l16)
        const float* wcol = W + (size_t)(half * 2) * N + col0 + l16;
#pragma unroll 8
        for (int k = 0; k < K; k += 4) {
            v2f a = *(const v2f*)(arow + k);      // offset k*4B      (immediate)
            v2f b;
            b.x = wcol[k * N];                     // offset k*N*4B    (immediate)
            b.y = wcol[k * N + N];                 // offset (k+1)*N*4B
            acc = __builtin_amdgcn_wmma_f32_16x16x4_f32(
                /*neg_a=*/false, a, /*neg_b=*/false, b,
                /*c_mod=*/(short)0, acc, /*reuse_a=*/false, /*reuse_b=*/false);
        }
    }

    float bias = 0.f;
    if (biasA) bias += biasA[col0 + l16];
    if (biasB) bias += biasB[col0 + l16];
    bias *= bias_scale;

    // C/D layout: VGPR v, lanes 0-15 -> M = v ; lanes 16-31 -> M = v + 8
#pragma unroll
    for (int v = 0; v < 8; ++v) {
        int m = row0 + v + half * 8;
        float val = out_scale * acc[v] + bias;
        if (do_relu) val = fmaxf(val, 0.f);
        C[(size_t)m * N + col0 + l16] = val;
    }
}

// ----------------------------- utility kernels -----------------------------
__global__ void zero_f32(float* __restrict__ p, long long n)
{
    long long i = (long long)blockIdx.x * blockDim.x + threadIdx.x;
    if (i < n) p[i] = 0.f;
}

// one block per edge, 256 threads = feature dims; agg[dst] += x[src]
__global__ __launch_bounds__(256) void scatter_add_feat(
    const float* __restrict__ x, const int* __restrict__ ei,
    float* __restrict__ agg)
{
    int e   = blockIdx.x;
    int h   = threadIdx.x;
    int src = ei[e];
    int dst = ei[kE + e];
    atomicAdd(&agg[(size_t)dst * kH + h], x[(size_t)src * kH + h]);
}

__global__ void count_edges(const int* __restrict__ ei, float* __restrict__ cnt)
{
    int e = blockIdx.x * blockDim.x + threadIdx.x;
    if (e < kE) atomicAdd(&cnt[ei[kE + e]], 1.0f);
}

__global__ __launch_bounds__(256) void normalize_rows(
    float* __restrict__ m, const float* __restrict__ cnt)
{
    int i = blockIdx.x, h = threadIdx.x;
    float c = fmaxf(cnt[i], 1.0f);
    m[(size_t)i * kH + h] *= (1.0f / c);
}

__global__ void count_batch(const int* __restrict__ batch, int n,
                            float* __restrict__ cnt)
{
    int i = blockIdx.x * blockDim.x + threadIdx.x;
    if (i < n) atomicAdd(&cnt[batch[i]], 1.0f);
}

__global__ __launch_bounds__(256) void pool_add(
    const float* __restrict__ x, const int* __restrict__ batch,
    float* __restrict__ pooled)
{
    int i = blockIdx.x, h = threadIdx.x;
    int b = batch[i];
    atomicAdd(&pooled[(size_t)b * kH + h], x[(size_t)i * kH + h]);
}

__global__ __launch_bounds__(256) void combine_pool(
    const float* __restrict__ p0, const float* __restrict__ p1,
    const float* __restrict__ p2,
    const float* __restrict__ c0, const float* __restrict__ c1,
    const float* __restrict__ c2,
    float* __restrict__ emb)
{
    int b = blockIdx.x, h = threadIdx.x;
    float v0 = p0[(size_t)b * kH + h] / fmaxf(c0[b], 1.f);
    float v1 = p1[(size_t)b * kH + h] / fmaxf(c1[b], 1.f);
    float v2 = p2[(size_t)b * kH + h] / fmaxf(c2[b], 1.f);
    emb[(size_t)b * kH + h] = (v0 + v1 + v2) * (1.0f / 3.0f);
}

// ------------------------------- launch ------------------------------------
extern "C" void kernel_launch(void* const* d_in, const int* in_sizes, int n_in,
                              void* d_out, int out_size, void* d_ws, size_t ws_size,
                              hipStream_t stream)
{
    const float* x_c  = (const float*)d_in[0];
    const float* x_1  = (const float*)d_in[1];
    const float* x_2  = (const float*)d_in[2];
    const int* ei0 = (const int*)d_in[3];   // central -> child_cont   (rel 0)
    const int* ei1 = (const int*)d_in[4];   // child_cont -> central   (rel 1)
    const int* ei2 = (const int*)d_in[5];   // central -> child_categ  (rel 2)
    const int* ei3 = (const int*)d_in[6];   // child_categ -> central  (rel 3)
    const int* b_c = (const int*)d_in[7];
    const int* b_1 = (const int*)d_in[8];
    const int* b_2 = (const int*)d_in[9];
    // d_in[10] = num_graphs (compile-time kB)
    const float* Wp_c = (const float*)d_in[11];
    const float* bp_c = (const float*)d_in[12];
    const float* Wp_1 = (const float*)d_in[13];
    const float* bp_1 = (const float*)d_in[14];
    const float* Wp_2 = (const float*)d_in[15];
    const float* bp_2 = (const float*)d_in[16];
    const float* Wmsg = (const float*)d_in[17];  // (2,4,H,H)
    const float* bmsg = (const float*)d_in[18];  // (2,4,H)
    const float* Wrt  = (const float*)d_in[19];  // (2,4,H,H)
    const float* Wg   = (const float*)d_in[20];  // (H,OUT)
    const float* bg   = (const float*)d_in[21];  // (OUT)

    // ---------------- workspace carve-out (floats) ----------------
    float* ws = (float*)d_ws;
    size_t off = 0;
    auto alloc = [&](size_t n) { float* p = ws + off; off += n; return p; };
    float* hc[2] = { alloc((size_t)kNC * kH), alloc((size_t)kNC * kH) };
    float* h1[2] = { alloc((size_t)kN1 * kH), alloc((size_t)kN1 * kH) };
    float* h2[2] = { alloc((size_t)kN2 * kH), alloc((size_t)kN2 * kH) };
    float* mca  = alloc((size_t)kNC * kH);   // mean buf: rel 1 -> central
    float* mcb  = alloc((size_t)kNC * kH);   // mean buf: rel 3 -> central
    float* mbig = alloc((size_t)kN1 * kH);   // reused: rel 0 then rel 2 (kN1==kN2)
    float* deg0 = alloc(kN1);
    float* deg1 = alloc(kNC);
    float* deg2 = alloc(kN2);
    float* deg3 = alloc(kNC);
    float* cb0  = alloc(kB);
    float* cb1  = alloc(kB);
    float* cb2  = alloc(kB);
    float* pool = alloc(3 * (size_t)kB * kH);
    float* emb  = alloc((size_t)kB * kH);

    auto zero = [&](float* p, long long n) {
        zero_f32<<<(unsigned)((n + 255) / 256), 256, 0, stream>>>(p, n);
    };

    // ---------------- degrees + batch counts (graph-structure constants) ----
    zero(deg0, kN1); zero(deg1, kNC); zero(deg2, kN2); zero(deg3, kNC);
    zero(cb0, kB);   zero(cb1, kB);   zero(cb2, kB);
    count_edges<<<(kE + 255) / 256, 256, 0, stream>>>(ei0, deg0);
    count_edges<<<(kE + 255) / 256, 256, 0, stream>>>(ei1, deg1);
    count_edges<<<(kE + 255) / 256, 256, 0, stream>>>(ei2, deg2);
    count_edges<<<(kE + 255) / 256, 256, 0, stream>>>(ei3, deg3);
    count_batch<<<(kNC + 255) / 256, 256, 0, stream>>>(b_c, kNC, cb0);
    count_batch<<<(kN1 + 255) / 256, 256, 0, stream>>>(b_1, kN1, cb1);
    count_batch<<<(kN2 + 255) / 256, 256, 0, stream>>>(b_2, kN2, cb2);

    // ---------------- input projections (WMMA) ----------------
    // central: K=64, N=256
    gemm_wmma_f32<64, 256><<<dim3(kNC / 16, 2), 256, 0, stream>>>(
        x_c, Wp_c, nullptr, nullptr, nullptr, nullptr, nullptr, nullptr,
        bp_c, nullptr, 1.f, 1.f, 0, hc[0]);
    // children: K=16, N=256
    gemm_wmma_f32<16, 256><<<dim3(kN1 / 16, 2), 256, 0, stream>>>(
        x_1, Wp_1, nullptr, nullptr, nullptr, nullptr, nullptr, nullptr,
        bp_1, nullptr, 1.f, 1.f, 0, h1[0]);
    gemm_wmma_f32<16, 256><<<dim3(kN2 / 16, 2), 256, 0, stream>>>(
        x_2, Wp_2, nullptr, nullptr, nullptr, nullptr, nullptr, nullptr,
        bp_2, nullptr, 1.f, 1.f, 0, h2[0]);

    // ---------------- 2 GNN layers ----------------
    const size_t WS = (size_t)kH * kH;  // one weight matrix
    int cur = 0;
    for (int l = 0; l < 2; ++l) {
        int nxt = cur ^ 1;
        const float* Wm = Wmsg + (size_t)l * 4 * WS;
        const float* Wr = Wrt  + (size_t)l * 4 * WS;
        const float* bm = bmsg + (size_t)l * 4 * kH;
        int relu = (l == 0);

        // central <- mean over rel1 (cont->c) and rel3 (categ->c)
        zero(mca, (long long)kNC * kH);
        zero(mcb, (long long)kNC * kH);
        scatter_add_feat<<<kE, 256, 0, stream>>>(h1[cur], ei1, mca);
        scatter_add_feat<<<kE, 256, 0, stream>>>(h2[cur], ei3, mcb);
        normalize_rows<<<kNC, 256, 0, stream>>>(mca, deg1);
        normalize_rows<<<kNC, 256, 0, stream>>>(mcb, deg3);
        gemm_wmma_f32<256, 256><<<dim3(kNC / 16, 2), 256, 0, stream>>>(
            mca, Wm + 1 * WS, mcb, Wm + 3 * WS,
            hc[cur], Wr + 1 * WS, hc[cur], Wr + 3 * WS,
            bm + 1 * kH, bm + 3 * kH, 0.5f, 0.5f, relu, hc[nxt]);

        // child_cont <- rel0 (central -> cont)
        zero(mbig, (long long)kN1 * kH);
        scatter_add_feat<<<kE, 256, 0, stream>>>(hc[cur], ei0, mbig);
        normalize_rows<<<kN1, 256, 0, stream>>>(mbig, deg0);
        gemm_wmma_f32<256, 256><<<dim3(kN1 / 16, 2), 256, 0, stream>>>(
            mbig, Wm + 0 * WS, h1[cur], Wr + 0 * WS,
            nullptr, nullptr, nullptr, nullptr,
            bm + 0 * kH, nullptr, 1.f, 1.f, relu, h1[nxt]);

        // child_categ <- rel2 (central -> categ)
        zero(mbig, (long long)kN2 * kH);
        scatter_add_feat<<<kE, 256, 0, stream>>>(hc[cur], ei2, mbig);
        normalize_rows<<<kN2, 256, 0, stream>>>(mbig, deg2);
        gemm_wmma_f32<256, 256><<<dim3(kN2 / 16, 2), 256, 0, stream>>>(
            mbig, Wm + 2 * WS, h2[cur], Wr + 2 * WS,
            nullptr, nullptr, nullptr, nullptr,
            bm + 2 * kH, nullptr, 1.f, 1.f, relu, h2[nxt]);

        cur = nxt;
    }

    // ---------------- global mean pool + fuse + final linear ----------------
    zero(pool, 3LL * kB * kH);
    pool_add<<<kNC, 256, 0, stream>>>(hc[cur], b_c, pool);
    pool_add<<<kN1, 256, 0, stream>>>(h1[cur], b_1, pool + (size_t)kB * kH);
    pool_add<<<kN2, 256, 0, stream>>>(h2[cur], b_2, pool + 2 * (size_t)kB * kH);
    combine_pool<<<kB, 256, 0, stream>>>(pool, pool + (size_t)kB * kH,
                                         pool + 2 * (size_t)kB * kH,
                                         cb0, cb1, cb2, emb);

    // final: K=256, N=128
    gemm_wmma_f32<256, 128><<<dim3(kB / 16, 1), 256, 0, stream>>>(
        emb, Wg, nullptr, nullptr, nullptr, nullptr, nullptr, nullptr,
        bg, nullptr, 1.f, 1.f, 0, (float*)d_out);
}